// Block_12558484374099
// MI455X (gfx1250) — compile-verified
//
#include <hip/hip_runtime.h>
#include <math.h>

// ---------------------------------------------------------------------------
// ViT block forward for MI455X (gfx1250, wave32, WMMA f32 16x16x4)
// B=32, N=577, C=768, H=3072.  All GEMMs are A @ B^T (B stored [N,K]).
// ---------------------------------------------------------------------------

typedef float v2f __attribute__((ext_vector_type(2)));
typedef float v8f __attribute__((ext_vector_type(8)));

#define BB 32
#define NN 577
#define CC 768
#define HH 3072
#define ROWS (BB * NN)            // 18464
#define PB (NN * CC)              // 443136 per-batch y elements
#define SROW (NN * NN)            // 332929 per-batch score elements
#define X_SIZE (ROWS * CC)        // 14180352

// ---------------------------------------------------------------------------
// LayerNorm: one block per row, C elements, mean/var reduction in LDS
// ---------------------------------------------------------------------------
__global__ void ln_kernel(const float* __restrict__ x, const float* __restrict__ w,
                          const float* __restrict__ b, float* __restrict__ out, int C) {
    __shared__ float s1[256];
    __shared__ float s2[256];
    size_t row = blockIdx.x;
    const float* xr = x + row * (size_t)C;
    int tid = threadIdx.x;
    float s = 0.f, ss = 0.f;
    for (int i = tid; i < C; i += 256) {
        float v = xr[i];
        s += v;
        ss += v * v;
    }
    s1[tid] = s; s2[tid] = ss;
    __syncthreads();
    for (int off = 128; off > 0; off >>= 1) {
        if (tid < off) { s1[tid] += s1[tid + off]; s2[tid] += s2[tid + off]; }
        __syncthreads();
    }
    float mu = s1[0] / (float)C;
    float var = s2[0] / (float)C - mu * mu;
    float inv = rsqrtf(var + 1e-5f);
    float* orow = out + row * (size_t)C;
    for (int i = tid; i < C; i += 256) {
        orow[i] = (xr[i] - mu) * inv * w[i] + b[i];
    }
}

// ---------------------------------------------------------------------------
// Softmax over attention rows (length 577), also emits token_attn for row 0
// ---------------------------------------------------------------------------
__global__ void softmax_kernel(float* __restrict__ S, float* __restrict__ tok) {
    __shared__ float red[256];
    int row = blockIdx.x;               // 0..ROWS-1
    int bb = row / NN;
    int ii = row - bb * NN;
    float* sr = S + (size_t)bb * SROW + (size_t)ii * NN;
    int tid = threadIdx.x;

    float mx = -3.402823466e38f;
    for (int j = tid; j < NN; j += 256) mx = fmaxf(mx, sr[j]);
    red[tid] = mx;
    __syncthreads();
    for (int off = 128; off > 0; off >>= 1) {
        if (tid < off) red[tid] = fmaxf(red[tid], red[tid + off]);
        __syncthreads();
    }
    mx = red[0];
    __syncthreads();

    float sum = 0.f;
    for (int j = tid; j < NN; j += 256) {
        float e = expf(sr[j] - mx);
        sr[j] = e;
        sum += e;
    }
    red[tid] = sum;
    __syncthreads();
    for (int off = 128; off > 0; off >>= 1) {
        if (tid < off) red[tid] += red[tid + off];
        __syncthreads();
    }
    float inv = 1.0f / red[0];

    for (int j = tid; j < NN; j += 256) {
        float p = sr[j] * inv;
        sr[j] = p;
        if (ii == 0 && j >= 1) tok[(size_t)bb * (NN - 1) + (j - 1)] = p;
    }
}

// ---------------------------------------------------------------------------
// V transpose: vt[b][c][j] = qkv[(b*NN+j)*2304 + 1536 + c]   ([B, C, N])
// so attn @ V can run through the (fast, contiguous) B^T GEMM path.
// ---------------------------------------------------------------------------
__global__ void transpose_v(const float* __restrict__ qkv, float* __restrict__ vt) {
    int idx = blockIdx.x * 256 + threadIdx.x;
    if (idx >= BB * CC * NN) return;
    unsigned b = (unsigned)idx / (unsigned)(CC * NN);
    unsigned r = (unsigned)idx - b * (unsigned)(CC * NN);
    unsigned c = r / (unsigned)NN;
    unsigned j = r - c * (unsigned)NN;
    vt[idx] = qkv[((size_t)b * NN + j) * (3 * CC) + 2 * CC + c];
}

// ---------------------------------------------------------------------------
// transpose(0,2,1).reshape scramble:  yp[b][p] = y[b][(p%577)*768 + p/577]
// ---------------------------------------------------------------------------
__global__ void permute_kernel(const float* __restrict__ y, float* __restrict__ yp) {
    int idx = blockIdx.x * 256 + threadIdx.x;
    if (idx >= X_SIZE) return;
    unsigned b = (unsigned)idx / (unsigned)PB;
    unsigned p = (unsigned)idx - b * (unsigned)PB;
    unsigned n = p % (unsigned)NN;
    unsigned c = p / (unsigned)NN;
    yp[idx] = y[(size_t)b * PB + (size_t)n * CC + c];
}

// ---------------------------------------------------------------------------
// Batched fp32 WMMA GEMM, A @ B^T only:
//   C = act( alpha * (A[M,K] @ B[N,K]^T + bias) ) + resid
// Block: 256 threads = 8 waves; block tile 128(M) x 128(N);
// each wave: 32x64 register tile = 2x4 accumulators of V_WMMA_F32_16X16X4_F32.
// All operand loads are contiguous float2 -> global_load_b64.
// ---------------------------------------------------------------------------
__global__ void __launch_bounds__(256)
wmma_gemm_nt(const float* __restrict__ A, long long lda, long long strideA,
             const float* __restrict__ Bw, long long ldb, long long strideB,
             float* __restrict__ Cp, long long ldc, long long strideC,
             int M, int N, int K, float alpha,
             const float* __restrict__ bias,
             const float* __restrict__ resid,
             int act) {
    int lane = threadIdx.x & 31;
    int wave = threadIdx.x >> 5;
    int wm = wave & 3;        // 4 waves along M
    int wn = wave >> 2;       // 2 waves along N
    int batch = blockIdx.z;

    const float* Ab = A + (size_t)batch * strideA;
    const float* Bb = Bw + (size_t)batch * strideB;
    float* Cb = Cp + (size_t)batch * strideC;
    const float* Rb = resid ? resid + (size_t)batch * strideC : nullptr;

    int rowBase = blockIdx.y * 128 + wm * 32;
    int colBase = blockIdx.x * 128 + wn * 64;

    int lm = lane & 15;
    int kh = (lane >> 4) * 2;     // this lane's k offset within a K=4 fragment

    // clamped indices: OOB rows/cols read valid memory, results never stored
    int rowA0 = rowBase + lm;
    int rowA1 = rowBase + 16 + lm;
    int rowA0c = rowA0 < M ? rowA0 : M - 1;
    int rowA1c = rowA1 < M ? rowA1 : M - 1;
    int colB[4], colBc[4];
#pragma unroll
    for (int t = 0; t < 4; ++t) {
        colB[t] = colBase + t * 16 + lm;
        colBc[t] = colB[t] < N ? colB[t] : N - 1;
    }

    v8f acc[2][4];
#pragma unroll
    for (int im = 0; im < 2; ++im)
#pragma unroll
        for (int t = 0; t < 4; ++t)
            acc[im][t] = (v8f){0.f, 0.f, 0.f, 0.f, 0.f, 0.f, 0.f, 0.f};

    const float* pa0 = Ab + (size_t)rowA0c * lda + kh;
    const float* pa1 = Ab + (size_t)rowA1c * lda + kh;
    const float* pb0 = Bb + (size_t)colBc[0] * ldb + kh;
    const float* pb1 = Bb + (size_t)colBc[1] * ldb + kh;
    const float* pb2 = Bb + (size_t)colBc[2] * ldb + kh;
    const float* pb3 = Bb + (size_t)colBc[3] * ldb + kh;

    int K4 = K & ~3;
#pragma unroll 4
    for (int k0 = 0; k0 < K4; k0 += 4) {
        v2f a0, a1, bf0, bf1, bf2, bf3;
        a0.x = pa0[0]; a0.y = pa0[1];
        a1.x = pa1[0]; a1.y = pa1[1];
        bf0.x = pb0[0]; bf0.y = pb0[1];
        bf1.x = pb1[0]; bf1.y = pb1[1];
        bf2.x = pb2[0]; bf2.y = pb2[1];
        bf3.x = pb3[0]; bf3.y = pb3[1];
        acc[0][0] = __builtin_amdgcn_wmma_f32_16x16x4_f32(false, a0, false, bf0, (short)0, acc[0][0], false, false);
        acc[1][0] = __builtin_amdgcn_wmma_f32_16x16x4_f32(false, a1, false, bf0, (short)0, acc[1][0], false, false);
        acc[0][1] = __builtin_amdgcn_wmma_f32_16x16x4_f32(false, a0, false, bf1, (short)0, acc[0][1], false, false);
        acc[1][1] = __builtin_amdgcn_wmma_f32_16x16x4_f32(false, a1, false, bf1, (short)0, acc[1][1], false, false);
        acc[0][2] = __builtin_amdgcn_wmma_f32_16x16x4_f32(false, a0, false, bf2, (short)0, acc[0][2], false, false);
        acc[1][2] = __builtin_amdgcn_wmma_f32_16x16x4_f32(false, a1, false, bf2, (short)0, acc[1][2], false, false);
        acc[0][3] = __builtin_amdgcn_wmma_f32_16x16x4_f32(false, a0, false, bf3, (short)0, acc[0][3], false, false);
        acc[1][3] = __builtin_amdgcn_wmma_f32_16x16x4_f32(false, a1, false, bf3, (short)0, acc[1][3], false, false);
        pa0 += 4; pa1 += 4;
        pb0 += 4; pb1 += 4; pb2 += 4; pb3 += 4;
    }

    if (K4 < K) {   // K % 4 tail (K=577 for attn@V), zero-fill guarded
        int kk = K4 + kh;
        bool k0ok = kk < K, k1ok = (kk + 1) < K;
        v2f a0, a1, bf0, bf1, bf2, bf3;
        a0.x = k0ok ? pa0[0] : 0.f;  a0.y = k1ok ? pa0[1] : 0.f;
        a1.x = k0ok ? pa1[0] : 0.f;  a1.y = k1ok ? pa1[1] : 0.f;
        bf0.x = k0ok ? pb0[0] : 0.f; bf0.y = k1ok ? pb0[1] : 0.f;
        bf1.x = k0ok ? pb1[0] : 0.f; bf1.y = k1ok ? pb1[1] : 0.f;
        bf2.x = k0ok ? pb2[0] : 0.f; bf2.y = k1ok ? pb2[1] : 0.f;
        bf3.x = k0ok ? pb3[0] : 0.f; bf3.y = k1ok ? pb3[1] : 0.f;
        acc[0][0] = __builtin_amdgcn_wmma_f32_16x16x4_f32(false, a0, false, bf0, (short)0, acc[0][0], false, false);
        acc[1][0] = __builtin_amdgcn_wmma_f32_16x16x4_f32(false, a1, false, bf0, (short)0, acc[1][0], false, false);
        acc[0][1] = __builtin_amdgcn_wmma_f32_16x16x4_f32(false, a0, false, bf1, (short)0, acc[0][1], false, false);
        acc[1][1] = __builtin_amdgcn_wmma_f32_16x16x4_f32(false, a1, false, bf1, (short)0, acc[1][1], false, false);
        acc[0][2] = __builtin_amdgcn_wmma_f32_16x16x4_f32(false, a0, false, bf2, (short)0, acc[0][2], false, false);
        acc[1][2] = __builtin_amdgcn_wmma_f32_16x16x4_f32(false, a1, false, bf2, (short)0, acc[1][2], false, false);
        acc[0][3] = __builtin_amdgcn_wmma_f32_16x16x4_f32(false, a0, false, bf3, (short)0, acc[0][3], false, false);
        acc[1][3] = __builtin_amdgcn_wmma_f32_16x16x4_f32(false, a1, false, bf3, (short)0, acc[1][3], false, false);
    }

    // Epilogue. C/D layout: vgpr i -> row = base + i + 8*(lane>>4); col = lane&15
    int rowOff = (lane >> 4) * 8;
#pragma unroll
    for (int im = 0; im < 2; ++im) {
#pragma unroll
        for (int t = 0; t < 4; ++t) {
            int col = colB[t];
            if (col >= N) continue;
            float bv = bias ? bias[col] : 0.f;
#pragma unroll
            for (int i = 0; i < 8; ++i) {
                int row = rowBase + im * 16 + rowOff + i;
                if (row < M) {
                    float v = (acc[im][t][i] + bv) * alpha;
                    if (act == 1) v = 0.5f * v * (1.0f + erff(v * 0.70710678118654752f));
                    if (Rb) v += Rb[(size_t)row * ldc + col];
                    Cb[(size_t)row * ldc + col] = v;
                }
            }
        }
    }
}

// ---------------------------------------------------------------------------
// Host-side orchestration
// ---------------------------------------------------------------------------
extern "C" void kernel_launch(void* const* d_in, const int* in_sizes, int n_in,
                              void* d_out, int out_size, void* d_ws, size_t ws_size,
                              hipStream_t stream) {
    const float* x      = (const float*)d_in[0];
    const float* n1w    = (const float*)d_in[1];
    const float* n1b    = (const float*)d_in[2];
    const float* qkv_w  = (const float*)d_in[3];
    const float* proj_w = (const float*)d_in[4];
    const float* proj_b = (const float*)d_in[5];
    const float* n2w    = (const float*)d_in[6];
    const float* n2b    = (const float*)d_in[7];
    const float* fc1_w  = (const float*)d_in[8];
    const float* fc1_b  = (const float*)d_in[9];
    const float* fc2_w  = (const float*)d_in[10];
    const float* fc2_b  = (const float*)d_in[11];

    float* out = (float*)d_out;
    float* ws  = (float*)d_ws;

    // workspace layout (floats), with reuse
    const size_t SZ_H   = (size_t)ROWS * CC;        // 14,180,352
    const size_t SZ_QKV = (size_t)ROWS * 3 * CC;    // 42,541,056
    const size_t SZ_S   = (size_t)BB * SROW;        // 10,653,728
    float* h    = ws;                               // LN1 out; reused as y later
    float* qkvb = ws + SZ_H;                        // qkv; reused as fc1 out later
    float* Sb   = qkvb + SZ_QKV;                    // scores/attn
    float* yp   = Sb + SZ_S;                        // permuted y
    float* xres = yp + SZ_H;                        // 2*(proj out)
    float* mb   = xres + SZ_H;                      // LN2 out
    float* vt   = mb + SZ_H;                        // V transposed [B, C, N]
    float* y    = h;                                // reuse: h dead after qkv GEMM
    float* g    = qkvb;                             // reuse: qkv/S/yp dead by fc1

    dim3 blk(256);
    const int MT = (ROWS + 127) / 128;              // 145 M-tiles
    const long long sQKV = (long long)NN * 3 * CC;  // per-batch qkv stride
    const float scale = 1.0f / sqrtf((float)CC);

    // 1. LN1
    ln_kernel<<<ROWS, blk, 0, stream>>>(x, n1w, n1b, h, CC);

    // 2. qkv = h @ qkv_w^T           [18464 x 2304]
    wmma_gemm_nt<<<dim3(3 * CC / 128, MT, 1), blk, 0, stream>>>(
        h, CC, 0, qkv_w, CC, 0, qkvb, 3 * CC, 0,
        ROWS, 3 * CC, CC, 1.0f, nullptr, nullptr, 0);

    // 3. S = scale * q @ k^T         per batch [577 x 577]
    wmma_gemm_nt<<<dim3((NN + 127) / 128, (NN + 127) / 128, BB), blk, 0, stream>>>(
        qkvb, 3 * CC, sQKV, qkvb + CC, 3 * CC, sQKV, Sb, NN, SROW,
        NN, NN, CC, scale, nullptr, nullptr, 0);

    // 4. softmax rows + token_attn
    softmax_kernel<<<ROWS, blk, 0, stream>>>(Sb, out + X_SIZE);

    // 5. vt = V^T per batch  [B, C, N]
    transpose_v<<<(BB * CC * NN + 255) / 256, blk, 0, stream>>>(qkvb, vt);

    // 6. y = attn @ vt^T             per batch [577 x 768], K=577 (tail)
    wmma_gemm_nt<<<dim3(CC / 128, (NN + 127) / 128, BB), blk, 0, stream>>>(
        Sb, NN, SROW, vt, NN, (long long)(CC * NN), y, CC, (long long)PB,
        NN, CC, NN, 1.0f, nullptr, nullptr, 0);

    // 7. transpose-scramble
    permute_kernel<<<(X_SIZE + 255) / 256, blk, 0, stream>>>(y, yp);

    // 8. xres = 2 * (yp @ proj_w^T + proj_b)
    wmma_gemm_nt<<<dim3(CC / 128, MT, 1), blk, 0, stream>>>(
        yp, CC, 0, proj_w, CC, 0, xres, CC, 0,
        ROWS, CC, CC, 2.0f, proj_b, nullptr, 0);

    // 9. LN2
    ln_kernel<<<ROWS, blk, 0, stream>>>(xres, n2w, n2b, mb, CC);

    // 10. g = gelu(m @ fc1_w^T + fc1_b)   [18464 x 3072]
    wmma_gemm_nt<<<dim3(HH / 128, MT, 1), blk, 0, stream>>>(
        mb, CC, 0, fc1_w, CC, 0, g, HH, 0,
        ROWS, HH, CC, 1.0f, fc1_b, nullptr, 1);

    // 11. out = g @ fc2_w^T + fc2_b + xres
    wmma_gemm_nt<<<dim3(CC / 128, MT, 1), blk, 0, stream>>>(
        g, HH, 0, fc2_w, HH, 0, out, CC, 0,
        ROWS, CC, HH, 1.0f, fc2_b, xres, 0);
}